// GATModule_34273839022829
// MI455X (gfx1250) — compile-verified
//
#include <hip/hip_runtime.h>
#include <hip/hip_bf16.h>

// GAT over sliding windows, restructured:
//   z_all[T,64] = X[T,64] @ W_fc[64,64]          (WMMA f32 16x16x4)
//   out[t] = sum_u softmax_u(lrelu(el[t-15+u]+er[t])) * z_all[t-15+u] + bias
// Memory-bound (~100 MB total traffic @ 23.3 TB/s); FP32 kept end-to-end.

typedef __attribute__((ext_vector_type(2))) float v2f;
typedef __attribute__((ext_vector_type(4))) float v4f;
typedef __attribute__((ext_vector_type(8))) float v8f;

#define THREADS 256
#define WAVES   8        // 256 threads / wave32
#define XPAD    68       // X-tile LDS row stride (words): 16B aligned, conflict-free gathers
#define WSTRIDE 160      // W LDS k-pair row stride (words): 160 % 64 == 32 -> bank shift/row

// ---------------------------------------------------------------------------
// Kernel 1: z = X @ W_fc via V_WMMA_F32_16X16X4_F32, one 16-row tile per wave.
// ---------------------------------------------------------------------------
__global__ __launch_bounds__(THREADS)
void gat_fc_wmma(const float* __restrict__ X, const float* __restrict__ Wfc,
                 float* __restrict__ Z, int T) {
  __shared__ float wlds[32 * WSTRIDE];          // swizzled W_fc, 20.5 KB
  __shared__ float xlds[WAVES][16 * XPAD];      // per-wave X/D staging, 34.8 KB

  const int tid = threadIdx.x;
  // Stage W_fc swizzled: W[k][n] -> wlds[(k>>1)*WSTRIDE + 2n + (k&1)]
  // so a B fragment {W[k0][n], W[k0+1][n]} (k0 even) is one ds_load_b64.
  for (int e = tid; e < 64 * 64; e += THREADS) {
    int k = e >> 6, n = e & 63;
    wlds[(k >> 1) * WSTRIDE + n * 2 + (k & 1)] = Wfc[e];
  }
  __syncthreads();

  const int wave = tid >> 5;
  const int lane = tid & 31;
  const int m16  = lane & 15;   // M (A/D) or N (B/D) index
  const int h    = lane >> 4;   // half-wave selector (K-half / M-half)
  float* xw = &xlds[wave][0];

  const int nTiles = (T + 15) >> 4;
  const int stride = (int)gridDim.x * WAVES;
  for (int tile = (int)blockIdx.x * WAVES + wave; tile < nTiles; tile += stride) {
    // ---- stage 16x64 X tile into LDS with coalesced b128 loads ----
#pragma unroll
    for (int i = 0; i < 8; ++i) {
      int e   = i * 32 + lane;                  // float4 index 0..255
      int row = e >> 4;
      int c4  = e & 15;
      int g   = tile * 16 + row; if (g > T - 1) g = T - 1;
      v4f v = *(const v4f*)(X + (size_t)g * 64 + c4 * 4);
      *(v4f*)(xw + row * XPAD + c4 * 4) = v;
    }
    // speculative prefetch of next tile (global_prefetch_b8)
    __builtin_prefetch(X + (size_t)(tile + stride) * 1024 + lane * 32, 0, 0);

    // ---- A fragments: a[kk] = {X[m][4kk+2h], X[m][4kk+2h+1]} ----
    v2f a[16];
#pragma unroll
    for (int kk = 0; kk < 16; ++kk)
      a[kk] = *(const v2f*)(xw + m16 * XPAD + kk * 4 + 2 * h);

    v8f acc[4];
#pragma unroll
    for (int nt = 0; nt < 4; ++nt)
#pragma unroll
      for (int i = 0; i < 8; ++i) acc[nt][i] = 0.0f;

    // ---- 4 N-tiles x 16 K-steps of f32 WMMA ----
#pragma unroll
    for (int nt = 0; nt < 4; ++nt) {
      const float* wp = wlds + h * WSTRIDE + (nt * 16 + m16) * 2;
#pragma unroll
      for (int kk = 0; kk < 16; ++kk) {
        v2f b = *(const v2f*)(wp + 2 * kk * WSTRIDE);
        acc[nt] = __builtin_amdgcn_wmma_f32_16x16x4_f32(
            false, a[kk], false, b, (short)0, acc[nt], false, false);
      }
    }

    // ---- repack D through LDS so global z stores are coalesced b128 ----
#pragma unroll
    for (int nt = 0; nt < 4; ++nt)
#pragma unroll
      for (int i = 0; i < 8; ++i)
        xw[(i + 8 * h) * XPAD + nt * 16 + m16] = acc[nt][i];
    // same-wave DS ops are processed in order; no barrier needed
#pragma unroll
    for (int i = 0; i < 8; ++i) {
      int e = i * 32 + lane;
      int row = e >> 4, c4 = e & 15;
      int g = tile * 16 + row;
      if (g < T)
        *(v4f*)(Z + (size_t)g * 64 + c4 * 4) = *(const v4f*)(xw + row * XPAD + c4 * 4);
    }
  }
}

// ---------------------------------------------------------------------------
// Kernel 2: fused el/er + softmax + weighted sum. 128 t's per block; the 143
// needed z rows live in LDS (window overlap -> z read ~1.1x from HBM).
// ---------------------------------------------------------------------------
#define TB 128
#define NR (TB + 15)   // 143 rows

__global__ __launch_bounds__(THREADS)
void gat_attn(const float* __restrict__ Z,
              const float* __restrict__ attn_l, const float* __restrict__ attn_r,
              const float* __restrict__ bias, float* __restrict__ out, int T) {
  __shared__ float zl[NR * 64];   // 36.6 KB
  __shared__ float el[NR];
  __shared__ float er[NR];

  const int tid = threadIdx.x;
  const int t0  = (int)blockIdx.x * TB;

  // stage z rows t0-15 .. t0+TB-1 (clamped; left padding == row 0)
  for (int e = tid; e < NR * 16; e += THREADS) {
    int j = e >> 4, c4 = e & 15;
    int g = t0 - 15 + j;
    g = g < 0 ? 0 : (g >= T ? T - 1 : g);
    *(v4f*)(zl + j * 64 + c4 * 4) = *(const v4f*)(Z + (size_t)g * 64 + c4 * 4);
  }
  __syncthreads();

  // attention scores per row (scalar-broadcast attn vectors)
  if (tid < NR) {
    float sl = 0.f, sr = 0.f;
    const float* row = zl + tid * 64;
#pragma unroll 4
    for (int c = 0; c < 64; c += 4) {
      v4f v  = *(const v4f*)(row + c);
      v4f al = *(const v4f*)(attn_l + c);
      v4f ar = *(const v4f*)(attn_r + c);
      sl += v.x * al.x + v.y * al.y + v.z * al.z + v.w * al.w;
      sr += v.x * ar.x + v.y * ar.y + v.z * ar.z + v.w * ar.w;
    }
    el[tid] = sl;
    er[tid] = sr;
  }
  __syncthreads();

  const int wave = tid >> 5, lane = tid & 31;
  v2f bi = *(const v2f*)(bias + 2 * lane);

  for (int k = wave; k < TB; k += WAVES) {   // one t per wave
    int t = t0 + k;
    if (t >= T) break;                       // uniform per wave
    float rt = er[k + 15];
    float ev[16];
    float mx = -3.0e38f;
#pragma unroll
    for (int u = 0; u < 16; ++u) {
      float e = el[k + u] + rt;
      e = e >= 0.f ? e : 0.2f * e;           // leaky_relu, slope 0.2
      ev[u] = e;
      mx = fmaxf(mx, e);
    }
    float s = 0.f;
#pragma unroll
    for (int u = 0; u < 16; ++u) { ev[u] = __expf(ev[u] - mx); s += ev[u]; }
    float inv = 1.0f / s;

    v2f acc; acc.x = 0.f; acc.y = 0.f;       // 2 features per lane
#pragma unroll
    for (int u = 0; u < 16; ++u) {
      float a = ev[u] * inv;
      v2f zv = *(const v2f*)(zl + (k + u) * 64 + 2 * lane);
      acc.x += a * zv.x; acc.y += a * zv.y;
    }
    acc.x += bi.x; acc.y += bi.y;
    *(v2f*)(out + (size_t)t * 64 + 2 * lane) = acc;
  }
}

// ---------------------------------------------------------------------------
extern "C" void kernel_launch(void* const* d_in, const int* in_sizes, int n_in,
                              void* d_out, int out_size, void* d_ws, size_t ws_size,
                              hipStream_t stream) {
  const float* X    = (const float*)d_in[0];  // ori_feats [T,64]
  const float* Wfc  = (const float*)d_in[1];  // [64,64]
  const float* al   = (const float*)d_in[2];  // [64]
  const float* ar   = (const float*)d_in[3];  // [64]
  const float* bias = (const float*)d_in[4];  // [64]
  // d_in[5] = window (==16, hardcoded)
  const int T = in_sizes[0] / 64;

  float* Z   = (float*)d_ws;                  // z_all [T,64], 25.6 MB scratch
  float* out = (float*)d_out;                 // [T,1,64]

  const int nTiles  = (T + 15) >> 4;
  const int blocks1 = (nTiles + WAVES - 1) / WAVES;
  gat_fc_wmma<<<blocks1, THREADS, 0, stream>>>(X, Wfc, Z, T);

  const int blocks2 = (T + TB - 1) / TB;
  gat_attn<<<blocks2, THREADS, 0, stream>>>(Z, al, ar, bias, out, T);
}